// MarchingSquares_89421219103314
// MI455X (gfx1250) — compile-verified
//
#include <hip/hip_runtime.h>
#include <stdint.h>

// ---------------------------------------------------------------------------
// Marching squares, MI455X (gfx1250).
// Memory-bound: ~269MB output stores + 33.5MB input reads -> ~15-20us floor
// @ 23.3 TB/s. Zero matmul FLOPs -> WMMA does not apply. CDNA5 paths used:
//   * Tensor Data Mover (tensor_load_to_lds + s_wait_tensorcnt) stages the
//     two coarse rows each block needs into LDS in one DMA per block.
//   * Non-temporal global_store_b128 for the 269MB output stream (output >
//     192MB L2, written once, never re-read -> NT keeps the 33.5MB input
//     resident in L2 between the count and emit passes).
//   * Deterministic 3-phase compaction (counts -> scan -> rank/emit) that
//     reproduces the reference's stable-sort ordering; tail-only zeroing.
// ---------------------------------------------------------------------------

#define HW    1024                 // input image side
#define HC    1025                 // cell grid side (padded)
#define NB    8                    // batch
#define NKEY  (2*NB)               // 16 sort keys
#define NCNT  (NKEY*HC)            // per-(key,row) count entries = 16400
#define ROWS  (NKEY*HC*HC)         // total output rows = 16,810,000
#define OUTF  (ROWS*4)             // output floats before counts = 67,240,000

typedef unsigned int u32x4 __attribute__((ext_vector_type(4)));
typedef int          i32x4 __attribute__((ext_vector_type(4)));
typedef int          i32x8 __attribute__((ext_vector_type(8)));
typedef float        f32x4 __attribute__((ext_vector_type(4)));

// Corner order SW=0, SE=1, NE=2, NW=3 ; CORNERS[:,0] -> y offset, [:,1] -> x.
__device__ const float CORN[4][2] = {
    {0.5f, -0.5f}, {0.5f, 0.5f}, {-0.5f, 0.5f}, {-0.5f, -0.5f}};

__device__ const int FIRST_T[16][4] = {
    {0,0,0,0},
    {3,0,1,0},{0,1,2,1},{3,0,2,1},{1,2,3,2},
    {3,0,1,0},{0,1,3,2},{3,0,3,2},{2,3,0,3},
    {2,3,1,0},{0,1,2,1},{2,3,2,1},{1,2,0,3},
    {1,2,1,0},{0,1,0,3},{0,0,0,0}};
__device__ const int SECOND5_T[4]  = {1,2,3,2};   // code 5:  SE,NE,NW,NE
__device__ const int SECOND10_T[4] = {2,3,0,3};   // code 10: NE,NW,SW,NW

// ---------------------------------------------------------------------------
// TDM: load 2 coarse rows (gy0, gy0+1) x 1024 cols of image b into LDS.
// D# per CDNA5 ISA: group0 = count/lds/global/type=2,
// group1 = data_size=4B, tensor dims (clipped so OOB rows zero-fill and are
// never read), tile 1024x2, stride 1024. Tracked on TENSORcnt.
// ---------------------------------------------------------------------------
__device__ __forceinline__ void load_two_rows(const float* __restrict__ src,
                                              float* lds_tile, int b, int gy0) {
#if __has_builtin(__builtin_amdgcn_tensor_load_to_lds)
  if (threadIdx.x < 32) {   // one wave issues the DMA; EXEC is ignored by TDM
    unsigned long long ga = (unsigned long long)src +
        4ull * ((unsigned long long)b * HW * HW + (unsigned long long)gy0 * HW);
    unsigned lds_off = (unsigned)(unsigned long long)(void*)lds_tile; // low 32b = LDS byte offset

    u32x4 g0;
    g0[0] = 1u;                                   // count=1 valid descriptor
    g0[1] = lds_off;                              // lds_addr
    g0[2] = (unsigned)ga;                         // global_addr[31:0]
    g0[3] = ((unsigned)(ga >> 32) & 0x01FFFFFFu)  // global_addr[56:32]
            | (2u << 30);                         // type = 2 ("image")

    int td0 = HW;          // tensor_dim0 (x)
    int td1 = HW - gy0;    // tensor_dim1 (rows left; rows past edge zero-fill)
    i32x8 g1;
    g1[0] = (2 << 16);                                        // data_size = 4B
    g1[1] = (td0 & 0xFFFF) << 16;                             // tensor_dim0[15:0]
    g1[2] = ((td0 >> 16) & 0xFFFF) | ((td1 & 0xFFFF) << 16);  // td0 hi | td1 lo
    g1[3] = ((td1 >> 16) & 0xFFFF) | (HW << 16);              // td1 hi | tile_dim0=1024
    g1[4] = 2;                                                // tile_dim1=2, tile_dim2=0
    g1[5] = HW;                                               // tensor_dim0_stride = 1024
    g1[6] = 0;
    g1[7] = 0;

    i32x4 gz = {0, 0, 0, 0};
#if defined(__clang_major__) && (__clang_major__ >= 23)
    i32x8 gz8 = {0, 0, 0, 0, 0, 0, 0, 0};
    __builtin_amdgcn_tensor_load_to_lds(g0, g1, gz, gz, gz8, 0);
#else
    __builtin_amdgcn_tensor_load_to_lds(g0, g1, gz, gz, 0);
#endif
    __builtin_amdgcn_s_wait_tensorcnt(0);
  }
#else
  // Fallback: cooperative load (kept so the file compiles on any toolchain).
  for (int i = threadIdx.x; i < 2 * HW; i += 256) {
    int r = i >> 10, c = i & (HW - 1);
    int gr = gy0 + r; if (gr > HW - 1) gr = HW - 1;
    lds_tile[i] = src[(size_t)b * HW * HW + (size_t)gr * HW + c];
  }
#endif
}

__device__ __forceinline__ int cell_code(const float* r0, const float* r1, int x) {
  int colA = (x > 0) ? (x - 1) : 0;        // clamp(x-1, 0, 1023)
  int colB = (x < HW) ? x : (HW - 1);      // clamp(x,   0, 1023)
  float nw = r0[colA], ne = r0[colB];
  float sw = r1[colA], se = r1[colB];
  return (sw > 0.f ? 1 : 0) | (se > 0.f ? 2 : 0) |
         (ne > 0.f ? 4 : 0) | (nw > 0.f ? 8 : 0);
}

__device__ __forceinline__ f32x4 seg_row(const int* req, float y, float x) {
  f32x4 r;
  r[0] = y + 0.5f * (CORN[req[0]][0] + CORN[req[1]][0]);
  r[1] = x + 0.5f * (CORN[req[0]][1] + CORN[req[1]][1]);
  r[2] = y + 0.5f * (CORN[req[2]][0] + CORN[req[3]][0]);
  r[3] = x + 0.5f * (CORN[req[2]][1] + CORN[req[3]][1]);
  return r;
}

// Exclusive block scan of 256 values (8 wave32 waves). Caller syncs before reuse.
__device__ __forceinline__ unsigned block_excl_scan_u(unsigned v, unsigned* slds,
                                                      unsigned* blockTot) {
  int lane = threadIdx.x & 31, wave = threadIdx.x >> 5;
  unsigned incl = v;
  for (int d = 1; d < 32; d <<= 1) {
    unsigned n = __shfl_up(incl, d, 32);
    if (lane >= d) incl += n;
  }
  if (lane == 31) slds[wave] = incl;
  __syncthreads();
  if (wave == 0) {
    unsigned w = (lane < 8) ? slds[lane] : 0u;
    for (int d = 1; d < 8; d <<= 1) {
      unsigned n = __shfl_up(w, d, 32);
      if (lane >= d) w += n;
    }
    if (lane < 8) slds[lane] = w;
  }
  __syncthreads();
  unsigned base = wave ? slds[wave - 1] : 0u;
  *blockTot = slds[7];
  return base + incl - v;
}

// --------------------------- Pass 1: per-row counts -------------------------
__global__ void __launch_bounds__(256)
ms_count(const float* __restrict__ src, unsigned* __restrict__ cnt) {
  __shared__ float tile[2][HW];
  __shared__ unsigned wsum[8];
  int y = blockIdx.y, b = blockIdx.z;
  int gy0 = (y > 0) ? (y - 1) : 0;
  load_two_rows(src, &tile[0][0], b, gy0);
  __syncthreads();
  const float* r0 = tile[0];                                  // clamp(y-1) row
  const float* r1 = tile[(y >= 1 && y <= HW - 1) ? 1 : 0];    // clamp(y) row

  unsigned acc = 0;   // packed: valid1 | valid2<<16
  for (int x0 = 0; x0 < HC; x0 += 256) {
    int x = x0 + (int)threadIdx.x;
    if (x < HC) {
      int code = cell_code(r0, r1, x);
      if (code > 0 && code < 15) acc += 1u;
      if (code == 5 || code == 10) acc += (1u << 16);
    }
  }
  for (int d = 16; d > 0; d >>= 1) acc += __shfl_xor(acc, d, 32);
  int lane = threadIdx.x & 31, wave = threadIdx.x >> 5;
  if (lane == 0) wsum[wave] = acc;
  __syncthreads();
  if (threadIdx.x == 0) {
    unsigned t = 0;
    for (int i = 0; i < 8; ++i) t += wsum[i];
    cnt[(2 * b) * HC + y]     = t & 0xFFFFu;
    cnt[(2 * b + 1) * HC + y] = t >> 16;
  }
}

// --------------------------- Pass 2: global scan ----------------------------
__global__ void __launch_bounds__(256)
ms_scan(const unsigned* __restrict__ cnt, unsigned* __restrict__ prefix,
        unsigned* __restrict__ meta, int* __restrict__ counts_out) {
  __shared__ unsigned slds[8];
  __shared__ unsigned keyTot[NKEY];
  if (threadIdx.x < NKEY) keyTot[threadIdx.x] = 0u;
  __syncthreads();
  unsigned carry = 0;
  for (int base = 0; base < NCNT; base += 256) {
    int i = base + (int)threadIdx.x;
    unsigned v = (i < NCNT) ? cnt[i] : 0u;
    if (i < NCNT && v) atomicAdd(&keyTot[i / HC], v);
    unsigned tot;
    unsigned ex = block_excl_scan_u(v, slds, &tot);
    if (i < NCNT) prefix[i] = carry + ex;
    carry += tot;
    __syncthreads();
  }
  if (threadIdx.x == 0) { prefix[NCNT] = carry; meta[0] = carry; }
  __syncthreads();
  if (threadIdx.x < NB)
    counts_out[threadIdx.x] =
        (int)(keyTot[2 * threadIdx.x] + keyTot[2 * threadIdx.x + 1]);
}

// --------------------------- Pass 3: zero the tail --------------------------
__global__ void __launch_bounds__(256)
ms_zero_tail(f32x4* __restrict__ out4, const unsigned* __restrict__ meta) {
  unsigned total = meta[0];
  unsigned r = blockIdx.x * 256u + threadIdx.x;
  if (r >= total && r < (unsigned)ROWS) {
    f32x4 z = {0.f, 0.f, 0.f, 0.f};
    __builtin_nontemporal_store(z, &out4[r]);    // global_store_b128 ... nt
  }
}

// --------------------------- Pass 4: rank and emit --------------------------
__global__ void __launch_bounds__(256)
ms_emit(const float* __restrict__ src, const unsigned* __restrict__ prefix,
        f32x4* __restrict__ out4) {
  __shared__ float tile[2][HW];
  __shared__ unsigned slds[8];
  int y = blockIdx.y, b = blockIdx.z;
  int gy0 = (y > 0) ? (y - 1) : 0;
  load_two_rows(src, &tile[0][0], b, gy0);
  __syncthreads();
  const float* r0 = tile[0];
  const float* r1 = tile[(y >= 1 && y <= HW - 1) ? 1 : 0];

  unsigned run1 = prefix[(2 * b) * HC + y];
  unsigned run2 = prefix[(2 * b + 1) * HC + y];
  float fy = (float)y;

  for (int x0 = 0; x0 < HC; x0 += 256) {
    int x = x0 + (int)threadIdx.x;
    int code = 0; bool v1 = false, v2 = false;
    if (x < HC) {
      code = cell_code(r0, r1, x);
      v1 = (code > 0) && (code < 15);
      v2 = (code == 5) || (code == 10);
    }
    unsigned pv = (v1 ? 1u : 0u) | (v2 ? (1u << 16) : 0u);
    unsigned tot;
    unsigned ex = block_excl_scan_u(pv, slds, &tot);
    if (v1) {
      f32x4 row = seg_row(FIRST_T[code], fy, (float)x);
      __builtin_nontemporal_store(row, &out4[run1 + (ex & 0xFFFFu)]);
    }
    if (v2) {
      f32x4 row = seg_row((code == 5) ? SECOND5_T : SECOND10_T, fy, (float)x);
      __builtin_nontemporal_store(row, &out4[run2 + (ex >> 16)]);
    }
    run1 += tot & 0xFFFFu;
    run2 += tot >> 16;
    __syncthreads();
  }
}

// ---------------------------------------------------------------------------
extern "C" void kernel_launch(void* const* d_in, const int* in_sizes, int n_in,
                              void* d_out, int out_size, void* d_ws, size_t ws_size,
                              hipStream_t stream) {
  const float* coarse = (const float*)d_in[0];

  unsigned* cnt    = (unsigned*)d_ws;            // [NCNT]
  unsigned* prefix = cnt + NCNT;                 // [NCNT+1]
  unsigned* meta   = prefix + NCNT + 1;          // [1] total valid rows

  f32x4* out4      = (f32x4*)d_out;
  int*   counts_o  = ((int*)d_out) + OUTF;       // 8 per-batch counts (int bits)

  dim3 gridTiles(1, HC, NB);
  ms_count<<<gridTiles, 256, 0, stream>>>(coarse, cnt);
  ms_scan<<<1, 256, 0, stream>>>(cnt, prefix, meta, counts_o);
  ms_zero_tail<<<(ROWS + 255) / 256, 256, 0, stream>>>(out4, meta);
  ms_emit<<<gridTiles, 256, 0, stream>>>(coarse, prefix, out4);
}